// MultiHeadGlobalAttention_68547678044319
// MI455X (gfx1250) — compile-verified
//
#include <hip/hip_runtime.h>
#include <math.h>

// MultiHeadGlobalAttention, MI455X / gfx1250, wave32.
// Algebraically collapsed: the [C x H*C] projection is folded out of both the
// logit path (w2 = W . tune, b2 = bias . tune) and the output path (softmax
// weights sum to 1, so out = (sum_n a*V) @ W + bias). Memory-bound:
// ~270 MB HBM traffic (V read twice; 2nd read is L2-resident: 134MB < 192MB L2)
// => ~11.5 us floor at 23.3 TB/s. All matrix math on V_WMMA_F32_16X16X4_F32.

#define B_ 64
#define N_ 4096
#define C_ 128
#define H_ 4
#define HC_ 512
#define ALPHA 0.2f

typedef __attribute__((ext_vector_type(8))) float v8f;
typedef __attribute__((ext_vector_type(2))) float v2f;

// workspace layout (float offsets), total ~5.4 MB
#define OFF_W2P   0u                                   // [128][16] padded w2
#define OFF_B2    2048u                                // [4] (pad to 16)
#define OFF_LOG   2064u                                // [B][H][N] logits
#define OFF_MX    (OFF_LOG + (unsigned)(B_*H_*N_))     // [B*H] max
#define OFF_ISUM  (OFF_MX + 256u)                      // [B*H] 1/sum
#define OFF_PART  (OFF_ISUM + 256u)                    // [B][8][H][C] partial u
#define OFF_U     (OFF_PART + (unsigned)(B_*8*H_*C_))  // [B][H][C] u

__device__ __forceinline__ float neg_inf() { return -__builtin_inff(); }

// ---- K1: w2[c',h] = sum_c W[c', h*C+c]*t[h,c] (padded to 16 cols), b2[h] ----
__global__ void k_prep(const float* __restrict__ W, const float* __restrict__ bias,
                       const float* __restrict__ T, float* __restrict__ ws) {
  int idx = blockIdx.x * 256 + threadIdx.x;
  if (idx < 2048) {
    int k = idx >> 4, j = idx & 15;
    float v = 0.f;
    if (j < H_) {
      const float* wr = W + (size_t)k * HC_ + j * C_;
      const float* tr = T + j * C_;
      for (int c = 0; c < C_; ++c) v = fmaf(wr[c], tr[c], v);
    }
    ws[OFF_W2P + idx] = v;                     // cols 4..15 are zero padding
  } else if (idx < 2048 + H_) {
    int j = idx - 2048;
    float v = 0.f;
    for (int c = 0; c < C_; ++c) v = fmaf(bias[j * C_ + c], T[j * C_ + c], v);
    ws[OFF_B2 + j] = v;
  }
}

// ---- K2: logits[b,h,n] = V[b,n,:] . w2[:,h] + b2[h]; leaky-relu; mask ------
// One wave = 16 nodes, A = V tile (16x4 per step), B = w2 slice (4x16).
__global__ void k_logits(const float* __restrict__ V, const int* __restrict__ gs,
                         float* __restrict__ ws) {
  __shared__ float s_w2p[2048];
  const int tid = threadIdx.x;
  for (int i = tid; i < 2048; i += 128) s_w2p[i] = ws[OFF_W2P + i];
  __syncthreads();
  const int b = blockIdx.y;
  const int wave = tid >> 5, lane = tid & 31;
  const int l16 = lane & 15, hi = lane >> 4;
  const int n0 = blockIdx.x * 64 + wave * 16;
  const float* vrow = V + ((size_t)b * N_ + (n0 + l16)) * C_;
  v8f acc = {};
  for (int k0 = 0; k0 < C_; k0 += 4) {
    v2f a = *(const v2f*)(vrow + k0 + 2 * hi);          // A[M=l16][K=k0+2hi, +1]
    v2f bv;
    bv.x = s_w2p[(k0 + 2 * hi) * 16 + l16];             // B[K][N=l16]
    bv.y = s_w2p[(k0 + 1 + 2 * hi) * 16 + l16];
    acc = __builtin_amdgcn_wmma_f32_16x16x4_f32(false, a, false, bv,
                                                (short)0, acc, false, false);
  }
  const int limit = gs[b];
  const float bb = ws[OFF_B2 + (l16 & 3)];
  if (l16 < H_) {                                       // col l16 = head
    float* lrow = ws + OFF_LOG + ((size_t)b * H_ + l16) * N_;
    for (int r = 0; r < 8; ++r) {
      int node = n0 + r + 8 * hi;                       // D row = node-in-tile
      float x = acc[r] + bb;
      x = (x >= 0.f) ? x : ALPHA * x;
      lrow[node] = (node < limit) ? x : neg_inf();
    }
  }
}

// ---- K3: per-(b,h) segment softmax stats: max and 1/sum(exp) ---------------
__global__ void k_stats(float* __restrict__ ws) {
  __shared__ float red[256];
  const int bh = blockIdx.x;
  const int tid = threadIdx.x;
  const float* l = ws + OFF_LOG + (size_t)bh * N_;
  float m = neg_inf();
  for (int n = tid; n < N_; n += 256) m = fmaxf(m, l[n]);
  red[tid] = m; __syncthreads();
  for (int s = 128; s > 0; s >>= 1) {
    if (tid < s) red[tid] = fmaxf(red[tid], red[tid + s]);
    __syncthreads();
  }
  m = red[0]; __syncthreads();
  float sum = 0.f;
  for (int n = tid; n < N_; n += 256) {
    float x = l[n];
    sum += (x == neg_inf()) ? 0.f : __expf(x - m);
  }
  red[tid] = sum; __syncthreads();
  for (int s = 128; s > 0; s >>= 1) {
    if (tid < s) red[tid] += red[tid + s];
    __syncthreads();
  }
  if (tid == 0) { ws[OFF_MX + bh] = m; ws[OFF_ISUM + bh] = 1.f / red[0]; }
}

// ---- K4: partial u[b,h,c'] = sum_{n in chunk} p[b,n,h] * V[b,n,c'] ---------
// A = probs (16 rows: heads 0..3 real, 4..15 zero), B = V (4 nodes x 16 c').
__global__ void k_wsum(const float* __restrict__ V, float* __restrict__ ws) {
  __shared__ float s_p[H_ * 512];
  const int b = blockIdx.y, chunk = blockIdx.x;
  const int nb = chunk * 512;
  const int tid = threadIdx.x;
  for (int i = tid; i < H_ * 512; i += 256) {
    int h = i >> 9, n = i & 511;
    float x = ws[OFF_LOG + ((size_t)b * H_ + h) * N_ + nb + n];
    float m = ws[OFF_MX + b * H_ + h];
    float is = ws[OFF_ISUM + b * H_ + h];
    s_p[i] = (x == neg_inf()) ? 0.f : __expf(x - m) * is;
  }
  __syncthreads();
  const int wave = tid >> 5, lane = tid & 31;
  const int l16 = lane & 15, hi = lane >> 4;
  const int c0 = wave * 16;
  const float* vb = V + ((size_t)b * N_ + nb) * C_;
  v8f acc = {};
  for (int k0 = 0; k0 < 512; k0 += 4) {                 // K = nodes in chunk
    v2f a;
    if (l16 < H_) {
      a.x = s_p[l16 * 512 + k0 + 2 * hi];
      a.y = s_p[l16 * 512 + k0 + 1 + 2 * hi];
    } else { a.x = 0.f; a.y = 0.f; }                    // zero-padded rows
    v2f bv;
    bv.x = vb[(size_t)(k0 + 2 * hi) * C_ + c0 + l16];
    bv.y = vb[(size_t)(k0 + 1 + 2 * hi) * C_ + c0 + l16];
    acc = __builtin_amdgcn_wmma_f32_16x16x4_f32(false, a, false, bv,
                                                (short)0, acc, false, false);
  }
  if (hi == 0) {                                        // D rows 0..3 = heads
    float* pr = ws + OFF_PART + (size_t)(b * 8 + chunk) * (H_ * C_);
    for (int r = 0; r < H_; ++r) pr[r * C_ + c0 + l16] = acc[r];
  }
}

// ---- K5: reduce 8 chunk partials -> u ---------------------------------------
__global__ void k_reduce(float* __restrict__ ws) {
  int idx = blockIdx.x * 256 + threadIdx.x;             // < B*H*C = 32768
  int b = idx >> 9, rest = idx & 511;
  float s = 0.f;
  for (int ch = 0; ch < 8; ++ch)
    s += ws[OFF_PART + ((size_t)(b * 8 + ch) << 9) + rest];
  ws[OFF_U + idx] = s;
}

// ---- K6: out[b, h*C+c] = u[b,h,:] @ W[:, h*C+c] + bias ----------------------
// Per head: [64 x 128] @ [128 x 128]; block = 8 waves (one 16-col N-tile each).
__global__ void k_out(const float* __restrict__ W, const float* __restrict__ bias,
                      const float* __restrict__ U, float* __restrict__ out) {
  const int m0 = blockIdx.x * 16;                       // batch tile
  const int h  = blockIdx.y;
  const int tid = threadIdx.x;
  const int wave = tid >> 5, lane = tid & 31;
  const int l16 = lane & 15, hi = lane >> 4;
  const int n0 = wave * 16;
  const float* urow = U + (size_t)(m0 + l16) * HC_ + h * C_;
  const float* wcol = W + h * C_ + n0 + l16;
  v8f acc = {};
  for (int k0 = 0; k0 < C_; k0 += 4) {
    v2f a = *(const v2f*)(urow + k0 + 2 * hi);
    v2f bv;
    bv.x = wcol[(size_t)(k0 + 2 * hi) * HC_];
    bv.y = wcol[(size_t)(k0 + 1 + 2 * hi) * HC_];
    acc = __builtin_amdgcn_wmma_f32_16x16x4_f32(false, a, false, bv,
                                                (short)0, acc, false, false);
  }
  for (int r = 0; r < 8; ++r) {
    int row = m0 + r + 8 * hi;
    int col = h * C_ + n0 + l16;
    out[(size_t)row * HC_ + col] = acc[r] + bias[col];
  }
}

extern "C" void kernel_launch(void* const* d_in, const int* in_sizes, int n_in,
                              void* d_out, int out_size, void* d_ws, size_t ws_size,
                              hipStream_t stream) {
  (void)in_sizes; (void)n_in; (void)out_size; (void)ws_size;
  const float* V    = (const float*)d_in[0];   // [B,N,C]
  const int*   gs   = (const int*)  d_in[1];   // [B]
  const float* W    = (const float*)d_in[2];   // [C, H*C]
  const float* bias = (const float*)d_in[3];   // [H*C]
  const float* T    = (const float*)d_in[4];   // [1,H,C]
  float* out = (float*)d_out;                  // [B, H*C]
  float* ws  = (float*)d_ws;

  k_prep  <<<9,            256, 0, stream>>>(W, bias, T, ws);
  k_logits<<<dim3(N_/64, B_), 128, 0, stream>>>(V, gs, ws);
  k_stats <<<B_ * H_,      256, 0, stream>>>(ws);
  k_wsum  <<<dim3(8, B_),  256, 0, stream>>>(V, ws);
  k_reduce<<<(B_*H_*C_)/256, 256, 0, stream>>>(ws);
  k_out   <<<dim3(B_/16, H_), 256, 0, stream>>>(W, bias, ws + OFF_U, out);
}